// SharedLinearPReLUModel_24051816857684
// MI455X (gfx1250) — compile-verified
//
#include <hip/hip_runtime.h>

typedef __attribute__((ext_vector_type(2))) float v2f;
typedef __attribute__((ext_vector_type(8))) float v8f;

// ---------------- workspace layout (float offsets) ----------------
// WT1   @ 0       : 7  x [32x64] transposed layer-1 Wl          (14336)
// WcT1  @ 14336   : 3  x [32x64] combined layer-1 weights       (6144)
// bc1   @ 20480   : 3  x [64]    combined layer-1 biases        (192)
// WT2   @ 20672   : 7  x [64x64] transposed layer-2 Wl          (28672)
// WcT2  @ 49344   : 3  x [64x64] combined layer-2 weights       (12288)
// bc2   @ 61632   : 3  x [64]    combined layer-2 biases        (192)
// FEAT  @ 62208   : h1_pf, h1_gw, h1_sw, h2_gw, h2_sw, agg, cnt

// ============ weight prep: transposes + combined (Wr-sum + self Wl) ============
__global__ __launch_bounds__(256) void prep_weights(
    const float* __restrict__ W1l, const float* __restrict__ b1l,
    const float* __restrict__ W1r, const float* __restrict__ W2l,
    const float* __restrict__ b2l, const float* __restrict__ W2r,
    float* __restrict__ wb)
{
  const int id = blockIdx.x;
  const int t  = threadIdx.x;
  const int rCnt[3]    = {3, 2, 2};           // dest: 0=gw,1=pf,2=sw
  const int rIdx[3][3] = {{0,4,6},{1,3,0},{2,5,0}};
  const int sIdx[3]    = {7, 9, 8};           // self-edge weight index per dest
  if (id < 7) {                               // layer-1 Wl transpose, K=32
    const float* in = W1l + id * 2048;
    float* out = wb + 0 + id * 2048;
    for (int i = t; i < 2048; i += 256) { int n = i >> 5, k = i & 31; out[k*64 + n] = in[i]; }
  } else if (id < 10) {                       // layer-1 combined
    int tt = id - 7;
    float* out = wb + 14336 + tt * 2048;
    float* bo  = wb + 20480 + tt * 64;
    int s = sIdx[tt];
    for (int i = t; i < 2048; i += 256) {
      int n = i >> 5, k = i & 31;
      float v = W1r[s*2048 + i] + W1l[s*2048 + i];
      for (int j = 0; j < rCnt[tt]; j++) v += W1r[rIdx[tt][j]*2048 + i];
      out[k*64 + n] = v;
    }
    if (t < 64) {
      float b = b1l[s*64 + t];
      for (int j = 0; j < rCnt[tt]; j++) b += b1l[rIdx[tt][j]*64 + t];
      bo[t] = b;
    }
  } else if (id < 17) {                       // layer-2 Wl transpose, K=64
    int e = id - 10;
    const float* in = W2l + e * 4096;
    float* out = wb + 20672 + e * 4096;
    for (int i = t; i < 4096; i += 256) { int n = i >> 6, k = i & 63; out[k*64 + n] = in[i]; }
  } else {                                    // layer-2 combined
    int tt = id - 17;
    float* out = wb + 49344 + tt * 4096;
    float* bo  = wb + 61632 + tt * 64;
    int s = sIdx[tt];
    for (int i = t; i < 4096; i += 256) {
      int n = i >> 6, k = i & 63;
      float v = W2r[s*4096 + i] + W2l[s*4096 + i];
      for (int j = 0; j < rCnt[tt]; j++) v += W2r[rIdx[tt][j]*4096 + i];
      out[k*64 + n] = v;
    }
    if (t < 64) {
      float b = b2l[s*64 + t];
      for (int j = 0; j < rCnt[tt]; j++) b += b2l[rIdx[tt][j]*64 + t];
      bo[t] = b;
    }
  }
}

// ============ edge scatter: wave-coalesced gather + HW float atomics ============
template<int K>
__global__ __launch_bounds__(256) void scatter_mean(
    const int* __restrict__ src, const int* __restrict__ dst,
    const float* __restrict__ X, float* __restrict__ agg,
    float* __restrict__ cnt, int E)
{
  const int TPE = K / 2;                      // threads per edge (2 floats each)
  long long gid = (long long)blockIdx.x * 256 + threadIdx.x;
  long long e = gid / TPE;
  if (e >= E) return;
  int p = (int)(gid - e * TPE);
  int s = src[e], d = dst[e];
  v2f v = *(const v2f*)(X + (size_t)s * K + p * 2);
  float* ap = agg + (size_t)d * K + p * 2;
  unsafeAtomicAdd(ap,     v.x);
  unsafeAtomicAdd(ap + 1, v.y);
  if (p == 0) unsafeAtomicAdd(cnt + d, 1.0f);
}

// ============ WMMA f32 GEMM ============
// ACCUM=0 : C[N,64]  = A[N,K] @ BT[K,64] + bias
// ACCUM=1 : C[N,64] += (A[N,K] * 1/max(cnt,1)) @ BT[K,64]
// Block = 128 threads = 4 wave32; block covers 16 rows; each wave one 16x16 C tile.
template<int K, bool ACCUM>
__global__ __launch_bounds__(128) void gemm_wmma(
    const float* __restrict__ A, const float* __restrict__ cnt,
    const float* __restrict__ BT, const float* __restrict__ bias,
    float* __restrict__ C, int N)
{
  constexpr int LDK = K + 4;                  // pad: 16B-aligned rows, no bank conflicts
  __shared__ float As[16 * LDK];
  __shared__ float scl[16];
  const int row0 = blockIdx.x * 16;
  const int t = threadIdx.x;

  if (ACCUM) {
    if (t < 16) {
      int row = row0 + t;
      float c = (row < N) ? cnt[row] : 1.f;
      scl[t] = 1.f / fmaxf(c, 1.f);
    }
    __syncthreads();
  }
  // stage A tile with float4 (b128) coalesced loads, mean scaling folded in
  for (int i = t * 4; i < 16 * K; i += 128 * 4) {
    int r = i / K, c = i - r * K;
    int row = row0 + r;
    float4 v = make_float4(0.f, 0.f, 0.f, 0.f);
    if (row < N) {
      v = *(const float4*)(A + (size_t)row * K + c);
      if (ACCUM) { float s = scl[r]; v.x *= s; v.y *= s; v.z *= s; v.w *= s; }
    }
    *(float4*)(&As[r * LDK + c]) = v;
  }
  __syncthreads();

  const int wave = t >> 5;
  const int lane = t & 31;
  const int m    = lane & 15;                 // row (A) / col (B,C) within half-wave
  const int hi   = lane >> 4;                 // selects K-pair (A/B) and M+8 (C/D)
  const int col  = wave * 16 + m;
  v8f acc = {};
#if __has_builtin(__builtin_amdgcn_wmma_f32_16x16x4_f32)
  #pragma unroll
  for (int kk = 0; kk < K; kk += 4) {
    const int ka = kk + hi * 2;
    v2f a, b;
    a.x = As[m * LDK + ka];
    a.y = As[m * LDK + ka + 1];
    b.x = BT[(size_t)ka * 64 + col];
    b.y = BT[(size_t)(ka + 1) * 64 + col];
    acc = __builtin_amdgcn_wmma_f32_16x16x4_f32(false, a, false, b, (short)0, acc,
                                                false, false);
  }
#else
  // scalar fallback with identical C register layout
  #pragma unroll
  for (int r = 0; r < 8; r++) {
    float s = 0.f;
    for (int k = 0; k < K; k++) s += As[(r + hi * 8) * LDK + k] * BT[(size_t)k * 64 + col];
    acc[r] = s;
  }
#endif
  // C/D layout: VGPR r -> M = r + 8*hi, N = lane%16
  if (row0 + 16 <= N) {                       // full tile (always true: N % 16 == 0)
    if (ACCUM) {
      #pragma unroll
      for (int r = 0; r < 8; r++) {
        size_t idx = (size_t)(row0 + r + hi * 8) * 64 + col;
        C[idx] += acc[r];
      }
    } else {
      const float b = bias[col];
      #pragma unroll
      for (int r = 0; r < 8; r++) {
        size_t idx = (size_t)(row0 + r + hi * 8) * 64 + col;
        C[idx] = acc[r] + b;
      }
    }
  } else {                                    // ragged fallback
    const float b = ACCUM ? 0.f : bias[col];
    #pragma unroll
    for (int r = 0; r < 8; r++) {
      int row = row0 + r + hi * 8;
      if (row < N) {
        size_t idx = (size_t)row * 64 + col;
        if (ACCUM) C[idx] += acc[r];
        else       C[idx] = acc[r] + b;
      }
    }
  }
}

// ============ elementwise ReLU (float4 / b128) ============
__global__ __launch_bounds__(256) void relu4_kernel(float4* __restrict__ p, long long n4)
{
  long long i = (long long)blockIdx.x * 256 + threadIdx.x;
  if (i < n4) {
    float4 v = p[i];
    v.x = fmaxf(v.x, 0.f); v.y = fmaxf(v.y, 0.f);
    v.z = fmaxf(v.z, 0.f); v.w = fmaxf(v.w, 0.f);
    p[i] = v;
  }
}

// ============ head: out = prelu(relu(h) @ linW.T + b), one wave per row ============
__global__ __launch_bounds__(256) void head_kernel(
    const float* __restrict__ H, const float* __restrict__ linW,
    const float* __restrict__ linb, const float* __restrict__ pa,
    float* __restrict__ out, int N)
{
  long long gid = (long long)blockIdx.x * 256 + threadIdx.x;
  int row = (int)(gid >> 5);
  int lane = (int)(gid & 31);
  if (row >= N) return;
  const float* h = H + (size_t)row * 64;
  float v = fmaxf(h[lane], 0.f) * linW[lane] +
            fmaxf(h[lane + 32], 0.f) * linW[lane + 32];
  for (int off = 16; off; off >>= 1) v += __shfl_xor(v, off, 32);
  if (lane == 0) {
    float r = v + linb[0];
    out[row] = (r >= 0.f) ? r : pa[0] * r;
  }
}

extern "C" void kernel_launch(void* const* d_in, const int* in_sizes, int n_in,
                              void* d_out_v, int out_size, void* d_ws, size_t ws_size,
                              hipStream_t stream)
{
  const float* x_pf = (const float*)d_in[0];
  const float* x_gw = (const float*)d_in[1];
  const float* x_sw = (const float*)d_in[2];
  const float* W1l  = (const float*)d_in[3];
  const float* b1l  = (const float*)d_in[4];
  const float* W1r  = (const float*)d_in[5];
  const float* W2l  = (const float*)d_in[6];
  const float* b2l  = (const float*)d_in[7];
  const float* W2r  = (const float*)d_in[8];
  const float* linW = (const float*)d_in[9];
  const float* linb = (const float*)d_in[10];
  const float* pa   = (const float*)d_in[11];
  const int* ei[7]; int Ecnt[7];
  for (int i = 0; i < 7; i++) { ei[i] = (const int*)d_in[12 + i]; Ecnt[i] = in_sizes[12 + i] / 2; }
  // self edges d_in[19..21] folded algebraically into combined weights.

  const int n_pf = in_sizes[0] / 32, n_gw = in_sizes[1] / 32, n_sw = in_sizes[2] / 32;
  int Ns[3] = {n_pf, n_gw, n_sw};                 // node type: 0=pf,1=gw,2=sw
  const int esrc[7] = {0, 1, 0, 2, 2, 1, 1};
  const int edst[7] = {1, 0, 2, 0, 1, 2, 1};

  float* ws = (float*)d_ws;
  float* F = ws + 62208;
  float* h1_pf = F;
  float* h1_gw = h1_pf + (size_t)n_pf * 64;
  float* h1_sw = h1_gw + (size_t)n_gw * 64;
  float* h2_gw = h1_sw + (size_t)n_sw * 64;
  float* h2_sw = h2_gw + (size_t)n_gw * 64;
  float* agg   = h2_sw + (size_t)n_sw * 64;       // sized for largest dest (gw)
  float* cntb  = agg   + (size_t)n_gw * 64;

  float* out    = (float*)d_out_v;
  float* out_gw = out;
  float* out_sw = out + n_gw;
  float* h2_pf  = out + n_gw + n_sw;              // h_pf output region, written directly

  prep_weights<<<20, 256, 0, stream>>>(W1l, b1l, W1r, W2l, b2l, W2r, ws);

  // -------- layer 1 (K = 32) --------
  {
    const float* X[3] = {x_pf, x_gw, x_sw};
    float* H[3] = {h1_pf, h1_gw, h1_sw};
    const float* WT  = ws + 0;
    const float* WcT = ws + 14336;
    const float* bc  = ws + 20480;
    for (int t = 0; t < 3; t++)
      gemm_wmma<32, false><<<(Ns[t] + 15) / 16, 128, 0, stream>>>(
          X[t], nullptr, WcT + t * 2048, bc + t * 64, H[t], Ns[t]);
    for (int e = 0; e < 7; e++) {
      int dN = Ns[edst[e]];
      hipMemsetAsync(agg,  0, (size_t)dN * 32 * sizeof(float), stream);
      hipMemsetAsync(cntb, 0, (size_t)dN * sizeof(float), stream);
      long long thr = (long long)Ecnt[e] * 16;
      scatter_mean<32><<<(int)((thr + 255) / 256), 256, 0, stream>>>(
          ei[e], ei[e] + Ecnt[e], X[esrc[e]], agg, cntb, Ecnt[e]);
      gemm_wmma<32, true><<<(dN + 15) / 16, 128, 0, stream>>>(
          agg, cntb, WT + e * 2048, nullptr, H[edst[e]], dN);
    }
    for (int t = 0; t < 3; t++) {
      long long n4 = (long long)Ns[t] * 16;       // N*64/4
      relu4_kernel<<<(int)((n4 + 255) / 256), 256, 0, stream>>>((float4*)H[t], n4);
    }
  }
  // -------- layer 2 (K = 64) --------
  {
    const float* X[3] = {h1_pf, h1_gw, h1_sw};
    float* H[3] = {h2_pf, h2_gw, h2_sw};
    const float* WT  = ws + 20672;
    const float* WcT = ws + 49344;
    const float* bc  = ws + 61632;
    for (int t = 0; t < 3; t++)
      gemm_wmma<64, false><<<(Ns[t] + 15) / 16, 128, 0, stream>>>(
          X[t], nullptr, WcT + t * 4096, bc + t * 64, H[t], Ns[t]);
    for (int e = 0; e < 7; e++) {
      int dN = Ns[edst[e]];
      hipMemsetAsync(agg,  0, (size_t)dN * 64 * sizeof(float), stream);
      hipMemsetAsync(cntb, 0, (size_t)dN * sizeof(float), stream);
      long long thr = (long long)Ecnt[e] * 32;
      scatter_mean<64><<<(int)((thr + 255) / 256), 256, 0, stream>>>(
          ei[e], ei[e] + Ecnt[e], X[esrc[e]], agg, cntb, Ecnt[e]);
      gemm_wmma<64, true><<<(dN + 15) / 16, 128, 0, stream>>>(
          agg, cntb, WT + e * 4096, nullptr, H[edst[e]], dN);
    }
    // only h_pf is a raw output; gw/sw ReLU is fused into the head kernel
    long long n4 = (long long)n_pf * 16;
    relu4_kernel<<<(int)((n4 + 255) / 256), 256, 0, stream>>>((float4*)h2_pf, n4);
  }
  // -------- head (applies ReLU on load) --------
  head_kernel<<<(int)(((long long)n_gw * 32 + 255) / 256), 256, 0, stream>>>(
      h2_gw, linW, linb, pa, out_gw, n_gw);
  head_kernel<<<(int)(((long long)n_sw * 32 + 255) / 256), 256, 0, stream>>>(
      h2_sw, linW, linb, pa, out_sw, n_sw);
}